// transinvariant_mlp_19696720019886
// MI455X (gfx1250) — compile-verified
//
#include <hip/hip_runtime.h>
#include <hip/hip_bf16.h>
#include <math.h>

#define NB    8
#define NDATA 512
#define NDIM  2
#define NCH   3
#define NMIX  4
#define HD    128
#define NF    (NMIX + 1)          // 5 features into layer 1

#define PI2F    6.2831853071795864f
#define SQG     4.4428829381583662f   // sqrt(0.5) * 2*pi  -> folds 0.5*(2pi)^2 into the squared distance
#define ZITTERF 0.0001f

typedef float v2f __attribute__((ext_vector_type(2)));
typedef float v8f __attribute__((ext_vector_type(8)));

// ---------------------------------------------------------------------------
// Stage 1: fused spectral-mixture kernel + contraction over c.
//   feat[b,a,k,m]   = sum_c exp(-||ea-ec||^2_scaled) * cos(pa-pc) * yc[b,c,k]
//                     + ZITTER*yc[b,a,k]
//   feat[b,a,k,4]   = yc[b,a,k]
// One wave per 'a' row; lanes split the 512-long c loop; Kcc never hits HBM.
// ---------------------------------------------------------------------------
__global__ __launch_bounds__(256) void stage1_feature(
    const float* __restrict__ xc, const float* __restrict__ yc,
    const float* __restrict__ mu, const float* __restrict__ istd,
    float* __restrict__ feat)
{
  __shared__ float xcs[NDATA * NDIM * NCH];   // 12 KB
  __shared__ float ycs[NDATA * NCH];          //  6 KB

  const int b   = blockIdx.x >> 6;            // NDATA/8 = 64 blocks per batch
  const int a0  = (blockIdx.x & 63) * 8;      // 8 waves -> 8 'a' rows per block
  const int tid = threadIdx.x;

  for (int i = tid; i < NDATA * NDIM * NCH; i += 256)
    xcs[i] = xc[b * NDATA * NDIM * NCH + i];
  for (int i = tid; i < NDATA * NCH; i += 256)
    ycs[i] = yc[b * NDATA * NCH + i];
  __syncthreads();

  const int wave = tid >> 5;
  const int lane = tid & 31;
  const int a    = a0 + wave;

  // Prescaled mixture constants (broadcast; served by scalar/L2 cache).
  float si[NMIX][NDIM], pm[NMIX][NDIM];
#pragma unroll
  for (int m = 0; m < NMIX; ++m)
#pragma unroll
    for (int d = 0; d < NDIM; ++d) {
      si[m][d] = SQG  * istd[m * NDIM + d];
      pm[m][d] = PI2F * mu[m * NDIM + d];
    }

  float xa0[NCH], xa1[NCH];
#pragma unroll
  for (int k = 0; k < NCH; ++k) {
    xa0[k] = xcs[(a * NDIM + 0) * NCH + k];
    xa1[k] = xcs[(a * NDIM + 1) * NCH + k];
  }

  float ax0[NMIX][NCH], ax1[NMIX][NCH], pa[NMIX][NCH], acc[NMIX][NCH];
#pragma unroll
  for (int m = 0; m < NMIX; ++m)
#pragma unroll
    for (int k = 0; k < NCH; ++k) {
      ax0[m][k] = si[m][0] * xa0[k];
      ax1[m][k] = si[m][1] * xa1[k];
      pa [m][k] = pm[m][0] * xa0[k] + pm[m][1] * xa1[k];
      acc[m][k] = 0.0f;
    }

  for (int c = lane; c < NDATA; c += 32) {
    float x0[NCH], x1[NCH], yv[NCH];
#pragma unroll
    for (int k = 0; k < NCH; ++k) {
      x0[k] = xcs[(c * NDIM + 0) * NCH + k];
      x1[k] = xcs[(c * NDIM + 1) * NCH + k];
      yv[k] = ycs[c * NCH + k];
    }
#pragma unroll
    for (int m = 0; m < NMIX; ++m)
#pragma unroll
      for (int k = 0; k < NCH; ++k) {
        float d0 = ax0[m][k] - si[m][0] * x0[k];
        float d1 = ax1[m][k] - si[m][1] * x1[k];
        float s  = d0 * d0 + d1 * d1;
        float dp = pa[m][k] - (pm[m][0] * x0[k] + pm[m][1] * x1[k]);
        acc[m][k] += __expf(-s) * __cosf(dp) * yv[k];
      }
  }

  // wave32 tree reduction of the 12 accumulators
#pragma unroll
  for (int off = 16; off > 0; off >>= 1)
#pragma unroll
    for (int m = 0; m < NMIX; ++m)
#pragma unroll
      for (int k = 0; k < NCH; ++k)
        acc[m][k] += __shfl_xor(acc[m][k], off, 32);

  if (lane == 0) {
    float* fp = &feat[(size_t)((b * NDATA + a) * NCH) * NF];
#pragma unroll
    for (int k = 0; k < NCH; ++k) {
      float ya = ycs[a * NCH + k];
#pragma unroll
      for (int m = 0; m < NMIX; ++m)
        fp[k * NF + m] = acc[m][k] + ZITTERF * ya;   // + ZITTER * eye diagonal
      fp[k * NF + NMIX] = ya;                        // concat yc
    }
  }
}

// ---------------------------------------------------------------------------
// Stage 2: fused layer-1 (K=5) + ReLU + mean over a -> X (8 x 384)
// ---------------------------------------------------------------------------
__global__ __launch_bounds__(HD) void stage2_layer1_mean(
    const float* __restrict__ feat, const float* __restrict__ w1,
    const float* __restrict__ b1, float* __restrict__ X)
{
  const int b = blockIdx.x / NCH;
  const int k = blockIdx.x % NCH;
  const int h = threadIdx.x;

  float w[NF];
#pragma unroll
  for (int j = 0; j < NF; ++j) w[j] = w1[h * NF + j];
  const float bias = b1[h];

  float acc = 0.0f;
  for (int a = 0; a < NDATA; ++a) {
    const float* f = &feat[(size_t)((b * NDATA + a) * NCH + k) * NF];
    float s = bias;
#pragma unroll
    for (int j = 0; j < NF; ++j) s = fmaf(f[j], w[j], s);
    acc += fmaxf(s, 0.0f);
  }
  X[b * (NCH * HD) + k * HD + h] = acc * (1.0f / NDATA);
}

// ---------------------------------------------------------------------------
// Stage 3: WMMA fp32 MLP tail. One wave owns one 16-wide N tile.
// A layout (16x4 f32): lane<16 -> row=lane, K={k0,k0+1}; lane>=16 -> K={k0+2,k0+3}
// B layout (4x16 f32): lane<16 -> col=lane, K={k0,k0+1}; lane>=16 -> K={k0+2,k0+3}
// D layout:            VGPR r -> M = (lane>=16 ? 8 : 0) + r, N = lane&15
// ---------------------------------------------------------------------------
__device__ __forceinline__ v8f wmma_f32_layer(
    const float* __restrict__ Asrc, int ldA,     // LDS, row-major M x K
    const float* __restrict__ W,    int ldW,     // global, [N][K] (row-major fan_out x fan_in)
    int nbase, int K, int l16, int khalf)
{
  v8f acc = {};
  for (int k0 = 0; k0 < K; k0 += 4) {
    v2f A, B;
    const float* ap = &Asrc[l16 * ldA + k0 + 2 * khalf];
    A.x = ap[0]; A.y = ap[1];
    const float* bp = &W[(nbase + l16) * ldW + k0 + 2 * khalf];
    B.x = bp[0]; B.y = bp[1];
    acc = __builtin_amdgcn_wmma_f32_16x16x4_f32(
        false, A, false, B, (short)0, acc, false, false);
  }
  return acc;
}

__global__ __launch_bounds__(256) void stage3_mlp(
    const float* __restrict__ X,
    const float* __restrict__ w2, const float* __restrict__ b2,
    const float* __restrict__ w3, const float* __restrict__ b3,
    const float* __restrict__ w4, const float* __restrict__ b4,
    const float* __restrict__ w5, const float* __restrict__ b5,
    float* __restrict__ out)
{
  __shared__ float Xs[16 * (NCH * HD)];   // 24 KB, rows 8..15 zero-padded
  __shared__ float Ha[16 * HD];           //  8 KB
  __shared__ float Hb[16 * HD];           //  8 KB

  const int tid = threadIdx.x;
  for (int i = tid; i < 16 * NCH * HD; i += 256)
    Xs[i] = (i < NB * NCH * HD) ? X[i] : 0.0f;
  __syncthreads();

  const int wave  = tid >> 5;
  const int lane  = tid & 31;
  const int l16   = lane & 15;
  const int khalf = lane >> 4;
  const int nbase = wave * 16;

  // layer 2: (16 x 384) @ w2^T -> Ha
  {
    v8f acc = wmma_f32_layer(Xs, NCH * HD, w2, NCH * HD, nbase, NCH * HD, l16, khalf);
    float bias = b2[nbase + l16];
#pragma unroll
    for (int r = 0; r < 8; ++r)
      Ha[(khalf * 8 + r) * HD + nbase + l16] = fmaxf(acc[r] + bias, 0.0f);
  }
  __syncthreads();

  // layer 3: (16 x 128) @ w3^T -> Hb
  {
    v8f acc = wmma_f32_layer(Ha, HD, w3, HD, nbase, HD, l16, khalf);
    float bias = b3[nbase + l16];
#pragma unroll
    for (int r = 0; r < 8; ++r)
      Hb[(khalf * 8 + r) * HD + nbase + l16] = fmaxf(acc[r] + bias, 0.0f);
  }
  __syncthreads();

  // layer 4: (16 x 128) @ w4^T -> Ha
  {
    v8f acc = wmma_f32_layer(Hb, HD, w4, HD, nbase, HD, l16, khalf);
    float bias = b4[nbase + l16];
#pragma unroll
    for (int r = 0; r < 8; ++r)
      Ha[(khalf * 8 + r) * HD + nbase + l16] = fmaxf(acc[r] + bias, 0.0f);
  }
  __syncthreads();

  // layer 5: (16 x 128) @ w5^T, N padded 12 -> 16; wave 0 only (uniform branch,
  // EXEC stays all-ones inside the wave for the WMMAs).
  if (wave == 0) {
    v8f acc = {};
    for (int k0 = 0; k0 < HD; k0 += 4) {
      v2f A, B;
      const float* ap = &Ha[l16 * HD + k0 + 2 * khalf];
      A.x = ap[0]; A.y = ap[1];
      if (l16 < NMIX * NCH) {
        const float* bp = &w5[l16 * HD + k0 + 2 * khalf];
        B.x = bp[0]; B.y = bp[1];
      } else {
        B.x = 0.0f; B.y = 0.0f;
      }
      acc = __builtin_amdgcn_wmma_f32_16x16x4_f32(
          false, A, false, B, (short)0, acc, false, false);
    }
    if (khalf == 0 && l16 < NMIX * NCH) {   // rows 0..7 valid, cols 0..11 valid
      float bias = b5[l16];
#pragma unroll
      for (int r = 0; r < 8; ++r)
        out[r * (NMIX * NCH) + l16] = acc[r] + bias;
    }
  }
}

// ---------------------------------------------------------------------------
extern "C" void kernel_launch(void* const* d_in, const int* in_sizes, int n_in,
                              void* d_out, int out_size, void* d_ws, size_t ws_size,
                              hipStream_t stream)
{
  (void)in_sizes; (void)n_in; (void)out_size; (void)ws_size;
  const float* xc   = (const float*)d_in[0];
  const float* yc   = (const float*)d_in[1];
  const float* mu   = (const float*)d_in[2];
  const float* istd = (const float*)d_in[3];
  const float* w1   = (const float*)d_in[4];
  const float* b1   = (const float*)d_in[5];
  const float* w2   = (const float*)d_in[6];
  const float* b2   = (const float*)d_in[7];
  const float* w3   = (const float*)d_in[8];
  const float* b3   = (const float*)d_in[9];
  const float* w4   = (const float*)d_in[10];
  const float* b4   = (const float*)d_in[11];
  const float* w5   = (const float*)d_in[12];
  const float* b5   = (const float*)d_in[13];
  float* out = (float*)d_out;

  float* feat = (float*)d_ws;                          // 8*512*3*5 f32 = 240 KB
  float* X    = feat + (size_t)NB * NDATA * NCH * NF;  // 8*384 f32

  stage1_feature    <<<NB * (NDATA / 8), 256, 0, stream>>>(xc, yc, mu, istd, feat);
  stage2_layer1_mean<<<NB * NCH,         HD,  0, stream>>>(feat, w1, b1, X);
  stage3_mlp        <<<1,                256, 0, stream>>>(X, w2, b2, w3, b3,
                                                           w4, b4, w5, b5, out);
}